// Cif_64141041598996
// MI455X (gfx1250) — compile-verified
//
#include <hip/hip_runtime.h>
#include <math.h>

// Problem constants (match reference)
#define BB 16
#define TT 1024
#define DD 1280
#define UU 128
#define LDA 1284   // padded LDS row stride in floats: 1284 % 64 == 4 -> conflict-free
#define KPAIRS 640 // DD/2 K-row pairs per tap
#define MT 32      // tokens per workgroup (two 16-row M-tiles share B fragments)

typedef __attribute__((ext_vector_type(2))) float v2f;
typedef __attribute__((ext_vector_type(4))) float v4f;
typedef __attribute__((ext_vector_type(8))) float v8f;

// ---------------------------------------------------------------------------
// Repack conv_w [D(out), D(in), 3] -> pair-interleaved:
//   Wp[((tap*640 + p)*1280 + n)*2 + r] = conv_w[n, 2p+r, tap]
// so a lane's B fragment (two consecutive K-rows, one column) is ONE b64 load
// and 16 lanes cover 128 contiguous bytes.
// ---------------------------------------------------------------------------
__global__ void __launch_bounds__(256)
repack_w(const float* __restrict__ cw, float* __restrict__ wp) {
    int j = blockIdx.x * blockDim.x + threadIdx.x;
    if (j >= 3 * DD * DD) return;
    const int r   = j & 1;
    const int n   = (j >> 1) % DD;
    const int pr  = (j >> 1) / DD;
    const int p   = pr % KPAIRS;
    const int tap = pr / KPAIRS;
    const int i   = 2 * p + r;
    wp[j] = cw[((size_t)n * DD + i) * 3 + tap];
}

__global__ void __launch_bounds__(256)
zero_f32(float* __restrict__ p, int n) {
    int i = blockIdx.x * blockDim.x + threadIdx.x;
    if (i < n) p[i] = 0.0f;
}

// ---------------------------------------------------------------------------
// Fused: conv1d(k=3,pad=1) -> +bias -> ReLU -> dot(lin_w)+lin_b -> sigmoid
//        -> *mask  => raw alphas [B,T]
// One workgroup = 32 tokens (two 16-row M-tiles). 8 waves, each owning a
// contiguous 160-column span (10 N-tiles) => full D=1280 output channels.
// Each B fragment (one b64/lane) feeds TWO wmma (both M-tiles): halves the
// L2 weight stream per FLOP vs a 16-token tile.
// ---------------------------------------------------------------------------
__global__ void __launch_bounds__(256)
conv_alpha(const float* __restrict__ hidden,
           const float* __restrict__ wp,
           const float* __restrict__ conv_b,
           const float* __restrict__ lin_w,
           const float* __restrict__ lin_b,
           const float* __restrict__ mask,
           float* __restrict__ alphas_raw) {
    extern __shared__ float shA[];        // 34 * LDA floats (~175 KB of 320 KB)
    __shared__ float tokAcc[MT];

    const int wgid = blockIdx.x;          // 0 .. B*T/32 - 1
    const int b    = wgid >> 5;           // 32 tiles per batch
    const int tile = wgid & 31;
    const int t0   = tile * MT;
    const int tid  = threadIdx.x;

    // Stage 34 rows (tokens t0-1 .. t0+32, zeros at batch edges) as float4
    const float* hb = hidden + (size_t)b * TT * DD;
    const v4f zero4 = (v4f){};
    for (int r = 0; r < MT + 2; ++r) {
        const int t = t0 + r - 1;
        const bool valid = (t >= 0) && (t < TT);
        const v4f* src = (const v4f*)(hb + (size_t)t * DD);
        v4f* dst = (v4f*)&shA[r * LDA];
        for (int c = tid; c < DD / 4; c += 256)
            dst[c] = valid ? src[c] : zero4;
    }
    if (tid < MT) tokAcc[tid] = 0.0f;
    __syncthreads();

    const int lane = tid & 31;
    const int wv   = tid >> 5;            // wave id 0..7
    const int hf   = lane >> 4;           // half-wave
    const int mrow = lane & 15;           // token row (A) / channel col (B)
    const int ncol = wv * 160 + mrow;     // this lane's base output channel

    v8f accA[10], accB[10];
#pragma unroll
    for (int i = 0; i < 10; ++i) { accA[i] = (v8f){}; accB[i] = (v8f){}; }

    for (int tap = 0; tap < 3; ++tap) {
        const float* arow0 = &shA[(mrow + tap) * LDA];        // tokens 0..15
        const float* arow1 = &shA[(mrow + 16 + tap) * LDA];   // tokens 16..31
        // v2f-indexed: element n of row-pair q lives at (q*1280 + n)
        const v2f* wbase = (const v2f*)wp + ((size_t)tap * KPAIRS) * DD + ncol;
        for (int kk = 0; kk < DD; kk += 4) {
            // A frags: VGPR0 = K{kk+2hf}, VGPR1 = K{kk+2hf+1} (8B-aligned LDS)
            v2f a0 = *(const v2f*)&arow0[kk + 2 * hf];
            v2f a1 = *(const v2f*)&arow1[kk + 2 * hf];
            const v2f* wkp = wbase + (size_t)((kk >> 1) + hf) * DD;
            // keep the L2-resident weight stream warm (~160KB ahead)
            __builtin_prefetch((const void*)(wkp + 16 * DD), 0, 3);
#pragma unroll
            for (int i = 0; i < 10; ++i) {
                v2f bf = wkp[i * 16];     // one b64, coalesced across lanes
                accA[i] = __builtin_amdgcn_wmma_f32_16x16x4_f32(
                    false, a0, false, bf, (short)0, accA[i], false, false);
                accB[i] = __builtin_amdgcn_wmma_f32_16x16x4_f32(
                    false, a1, false, bf, (short)0, accB[i], false, false);
            }
        }
    }

    // Epilogue: bias + ReLU + lin_w dot, reduce over channels into 32 logits
    float sA[8], sB[8];
#pragma unroll
    for (int v = 0; v < 8; ++v) { sA[v] = 0.0f; sB[v] = 0.0f; }
#pragma unroll
    for (int i = 0; i < 10; ++i) {
        const int nn = ncol + i * 16;
        const float cb = conv_b[nn];
        const float lw = lin_w[nn];
#pragma unroll
        for (int v = 0; v < 8; ++v) {
            float rA = accA[i][v] + cb; rA = rA > 0.0f ? rA : 0.0f;
            float rB = accB[i][v] + cb; rB = rB > 0.0f ? rB : 0.0f;
            sA[v] = fmaf(rA, lw, sA[v]);
            sB[v] = fmaf(rB, lw, sB[v]);
        }
    }
    // C/D layout: VGPR v, lane -> token (v + 8*hf), channel mrow
#pragma unroll
    for (int v = 0; v < 8; ++v) {
        atomicAdd(&tokAcc[v + 8 * hf], sA[v]);
        atomicAdd(&tokAcc[16 + v + 8 * hf], sB[v]);
    }
    __syncthreads();

    if (tid < MT) {
        const int t = t0 + tid;
        float logit = tokAcc[tid] + lin_b[0];
        float alpha = 1.0f / (1.0f + expf(-logit));
        alpha *= mask[b * TT + t];            // mask is [B,1,T]
        alphas_raw[b * TT + t] = alpha;
    }
}

// ---------------------------------------------------------------------------
// token_num[b] = sum_t alphas_raw; alphas_out = alphas_raw * (tgt/token_num)
// ---------------------------------------------------------------------------
__global__ void __launch_bounds__(256)
normalize_k(const float* __restrict__ alphas_raw,
            const float* __restrict__ tgt_len,
            float* __restrict__ token_num_out,
            float* __restrict__ alphas_out) {
    __shared__ float red[256];
    __shared__ float scale_sh;
    const int b = blockIdx.x, tid = threadIdx.x;
    float s = 0.0f;
    for (int t = tid; t < TT; t += 256) s += alphas_raw[b * TT + t];
    red[tid] = s;
    __syncthreads();
    for (int off = 128; off > 0; off >>= 1) {
        if (tid < off) red[tid] += red[tid + off];
        __syncthreads();
    }
    if (tid == 0) {
        const float tok = red[0];
        token_num_out[b] = tok;
        scale_sh = tgt_len[b] / tok;
    }
    __syncthreads();
    const float scale = scale_sh;
    for (int t = tid; t < TT; t += 256)
        alphas_out[b * TT + t] = alphas_raw[b * TT + t] * scale;
}

// ---------------------------------------------------------------------------
// Sequential scalar CIF pass: per-t weights (cur, remainder, fire slot) and
// fires output. Only true serial dependency of the whole problem (~4 KFLOP).
// ---------------------------------------------------------------------------
__global__ void cif_weights(const float* __restrict__ alphas_norm,
                            float* __restrict__ fires_out,
                            float* __restrict__ cur_w,
                            float* __restrict__ rem_w,
                            int*   __restrict__ slot_w) {
    const int b = blockIdx.x;
    if (threadIdx.x != 0) return;
    float integ = 0.0f;
    int cnt = 0;
    for (int t = 0; t < TT; ++t) {
        const float alpha = alphas_norm[b * TT + t];
        const float dist  = 1.0f - integ;
        integ += alpha;
        fires_out[b * TT + t] = integ;
        if (integ >= 1.0f) {
            integ -= 1.0f;
            cur_w[b * TT + t]  = dist;
            rem_w[b * TT + t]  = alpha - dist;
            slot_w[b * TT + t] = cnt++;
        } else {
            cur_w[b * TT + t]  = alpha;
            rem_w[b * TT + t]  = 0.0f;
            slot_w[b * TT + t] = -1;
        }
    }
}

// ---------------------------------------------------------------------------
// Frame accumulation: weights known -> parallel across D, serial over T in
// registers. Writes only fired rows into (pre-zeroed) acoustic [B,U,D].
// ---------------------------------------------------------------------------
__global__ void __launch_bounds__(256)
cif_frames(const float* __restrict__ hidden,
           const float* __restrict__ cur_w,
           const float* __restrict__ rem_w,
           const int*   __restrict__ slot_w,
           float* __restrict__ acoustic) {
    const int b = blockIdx.x;
    const int d = blockIdx.y * 256 + threadIdx.x;   // gridDim.y = D/256
    const float* hb = hidden + (size_t)b * TT * DD + d;
    float acc = 0.0f;
    for (int t = 0; t < TT; ++t) {
        const float val = hb[(size_t)t * DD];
        const float c   = cur_w[b * TT + t];        // uniform -> s_load
        const int   sl  = slot_w[b * TT + t];
        const float tmp = fmaf(c, val, acc);
        if (sl >= 0) {
            if (sl < UU)
                acoustic[((size_t)b * UU + sl) * DD + d] = tmp;
            acc = rem_w[b * TT + t] * val;
        } else {
            acc = tmp;
        }
    }
}

// ---------------------------------------------------------------------------
extern "C" void kernel_launch(void* const* d_in, const int* in_sizes, int n_in,
                              void* d_out, int out_size, void* d_ws, size_t ws_size,
                              hipStream_t stream) {
    const float* hidden = (const float*)d_in[0];
    const float* mask   = (const float*)d_in[1];
    const float* tgt    = (const float*)d_in[2];
    const float* conv_w = (const float*)d_in[3];
    const float* conv_b = (const float*)d_in[4];
    const float* lin_w  = (const float*)d_in[5];
    const float* lin_b  = (const float*)d_in[6];

    // Output tuple layout (flat, return order):
    // acoustic [B,U,D] | token_num [B] | alphas [B,T] | fires [B,T]
    float* out        = (float*)d_out;
    float* acoustic   = out;
    float* token_num  = out + (size_t)BB * UU * DD;
    float* alphas_out = token_num + BB;
    float* fires_out  = alphas_out + BB * TT;

    // Workspace layout
    char* ws = (char*)d_ws;
    float* wpair      = (float*)ws;  ws += (size_t)3 * DD * DD * sizeof(float);
    float* alphas_raw = (float*)ws;  ws += (size_t)BB * TT * sizeof(float);
    float* cur_w      = (float*)ws;  ws += (size_t)BB * TT * sizeof(float);
    float* rem_w      = (float*)ws;  ws += (size_t)BB * TT * sizeof(float);
    int*   slot_w     = (int*)ws;

    repack_w<<<(3 * DD * DD + 255) / 256, 256, 0, stream>>>(conv_w, wpair);
    zero_f32<<<(BB * UU * DD + 255) / 256, 256, 0, stream>>>(acoustic, BB * UU * DD);

    const size_t shbytes = (size_t)(MT + 2) * LDA * sizeof(float); // ~175 KB / 320 KB WGP
    conv_alpha<<<BB * TT / MT, 256, shbytes, stream>>>(
        hidden, wpair, conv_b, lin_w, lin_b, mask, alphas_raw);

    normalize_k<<<BB, 256, 0, stream>>>(alphas_raw, tgt, token_num, alphas_out);
    cif_weights<<<BB, 32, 0, stream>>>(alphas_out, fires_out, cur_w, rem_w, slot_w);

    dim3 g3(BB, DD / 256);
    cif_frames<<<g3, 256, 0, stream>>>(hidden, cur_w, rem_w, slot_w, acoustic);
}